// GCN_3_layer_fc_45311904973171
// MI455X (gfx1250) — compile-verified
//
#include <hip/hip_runtime.h>
#include <hip/hip_bf16.h>

// ---------------------------------------------------------------------------
// GCN 3-layer + residual + output projection, CDNA5 (gfx1250, wave32, WMMA).
//
// Dense GEMMs use V_WMMA_F32_16X16X4_F32 (full fp32 precision, f32 acc).
// Each wave computes MT=5 independent 16x16 output tiles sharing one B
// fragment per k-step (B register reuse 5x, independent accumulators ->
// back-to-back XDL issue with no hazard NOPs). Weights are pre-transposed
// once so both A and B fragments are contiguous global_load_b64.
//
// Operand layouts per CDNA5 ISA 7.12.2:
//   A 16x4 (MxK): lane L holds row M=(L&15); lanes<16 carry K={k,k+1},
//                 lanes>=16 carry K={k+2,k+3}  -> v2f per lane
//   B 4x16 (KxN): mirrored with columns       -> v2f per lane
//   C/D 16x16:    VGPR r, lanes<16 -> M=r, lanes>=16 -> M=r+8, N=(L&15)
// ---------------------------------------------------------------------------

typedef __attribute__((ext_vector_type(2))) float v2f;
typedef __attribute__((ext_vector_type(8))) float v8f;

#define NODES 50000
#define EDGES 800000
#define FDIM  128     // F_IN == H == 128 (K of every GEMM)
#define CDIM  64
#define MTILE 5       // m-tiles per wave; 50000 = 625 blocks * (5*16) rows

// ---------------------------------------------------------------- utilities
__global__ void zero_f32_kernel(float* __restrict__ p, int n) {
  int i = blockIdx.x * blockDim.x + threadIdx.x;
  if (i < n) p[i] = 0.0f;
}

__global__ void degree_kernel(const int* __restrict__ src,
                              const int* __restrict__ dst,
                              float* __restrict__ deg_out,
                              float* __restrict__ deg_in, int e) {
  int i = blockIdx.x * blockDim.x + threadIdx.x;
  if (i < e) {
    atomicAdd(&deg_out[src[i]], 1.0f);   // global_atomic_add_f32 (no return)
    atomicAdd(&deg_in [dst[i]], 1.0f);
  }
}

__global__ void norm_kernel(float* __restrict__ d_src,
                            float* __restrict__ d_dst, int n) {
  int i = blockIdx.x * blockDim.x + threadIdx.x;
  if (i < n) {
    float a = d_src[i], b = d_dst[i];
    d_src[i] = rsqrtf(a > 0.0f ? a : 1.0f);   // deg^-1/2, zero-deg clamped to 1
    d_dst[i] = rsqrtf(b > 0.0f ? b : 1.0f);
  }
}

// WT[c, k] = W[k, c]  (makes B fragments contiguous b64 loads in the GEMM)
__global__ void transpose_w_kernel(const float* __restrict__ W,
                                   float* __restrict__ WT, int kdim, int ndim) {
  int i = blockIdx.x * blockDim.x + threadIdx.x;
  if (i < kdim * ndim) {
    int k = i / ndim, c = i % ndim;
    WT[(size_t)c * kdim + k] = W[i];
  }
}

// ------------------------------------------------------------- dense GEMM
// out[M,LDN] = (SCALE_A ? diag(dsrc)*A : A) @ W  (+ bias if ADD_BIAS)
// A is [nrows,128] row-major, WT is [LDN,128] (transposed W).
// blockDim.x = 32*(LDN/16); wave w -> column tile w; each wave does MTILE
// consecutive 16-row tiles. gridDim.x = nrows/(16*MTILE) (exact here).
template <int LDN, bool SCALE_A, bool ADD_BIAS>
__global__ void gemm_k128_wmma(const float* __restrict__ A,
                               const float* __restrict__ WT,
                               const float* __restrict__ bias,
                               const float* __restrict__ dsrc,
                               float* __restrict__ out) {
  const int lane = threadIdx.x & 31;
  const int wave = threadIdx.x >> 5;
  const int m0   = blockIdx.x * (16 * MTILE);  // uniform per wave
  const int n0   = wave << 4;                  // uniform per wave
  const int col  = n0 + (lane & 15);
  const int kb   = (lane >> 4) << 1;           // 0 (lanes 0-15) or 2 (16-31)

  float init = 0.0f;
  if constexpr (ADD_BIAS) init = bias[col];
  v8f acc[MTILE];
#pragma unroll
  for (int t = 0; t < MTILE; ++t)
#pragma unroll
    for (int r = 0; r < 8; ++r) acc[t][r] = init;

  const float* __restrict__ arow[MTILE];
  float scale[MTILE];
#pragma unroll
  for (int t = 0; t < MTILE; ++t) {
    const int row = m0 + t * 16 + (lane & 15);
    arow[t]  = A + (size_t)row * FDIM;
    scale[t] = SCALE_A ? dsrc[row] : 1.0f;
  }
  const float* __restrict__ bcol = WT + (size_t)col * FDIM;

#pragma unroll
  for (int k = 0; k < FDIM; k += 4) {          // 32 k-steps, 5 WMMA each
    v2f b;
    b.x = bcol[k + kb];                        // contiguous b64
    b.y = bcol[k + kb + 1];
#pragma unroll
    for (int t = 0; t < MTILE; ++t) {
      v2f a;
      a.x = arow[t][k + kb];                   // contiguous b64
      a.y = arow[t][k + kb + 1];
      if constexpr (SCALE_A) { a.x *= scale[t]; a.y *= scale[t]; }
      acc[t] = __builtin_amdgcn_wmma_f32_16x16x4_f32(
          /*neg_a=*/false, a, /*neg_b=*/false, b,
          /*c_mod=*/(short)0, acc[t], /*reuse_a=*/false, /*reuse_b=*/false);
    }
  }

  const int rbase = (lane >> 4) << 3;          // lanes 16-31 hold rows M=8..15
#pragma unroll
  for (int t = 0; t < MTILE; ++t)
#pragma unroll
    for (int r = 0; r < 8; ++r)
      out[(size_t)(m0 + t * 16 + rbase + r) * LDN + col] = acc[t][r];
}

// --------------------------------------------------------------- SpMM edge
// agg[dst[e], :] += h[src[e], :]   -- one edge per wave; lane handles 4 floats
__global__ void scatter_edges_kernel(const float* __restrict__ h,
                                     const int* __restrict__ src,
                                     const int* __restrict__ dst,
                                     float* __restrict__ agg, int e) {
  const int lane   = threadIdx.x & 31;
  const int gwave  = (blockIdx.x * blockDim.x + threadIdx.x) >> 5;
  const int nwaves = (gridDim.x * blockDim.x) >> 5;
  for (int i = gwave; i < e; i += nwaves) {
    const int s = src[i];
    const int d = dst[i];
    if (i + nwaves < e) {                  // hide the gather latency
      const int sn = src[i + nwaves];
      __builtin_prefetch(h + (size_t)sn * FDIM + lane * 4, 0, 1);
    }
    const float4 v =
        ((const float4*)(h + (size_t)s * FDIM))[lane];   // 512B/wave coalesced
    float* ap = agg + (size_t)d * FDIM + (lane << 2);
    atomicAdd(ap + 0, v.x);
    atomicAdd(ap + 1, v.y);
    atomicAdd(ap + 2, v.z);
    atomicAdd(ap + 3, v.w);
  }
}

// ---------------------------------------------------------------- epilogue
// out = [relu]( agg * d_dst[row] + bias[col] [+ res] )
template <bool RELU, bool ADD_RES>
__global__ void epilogue_kernel(const float* __restrict__ agg,
                                const float* __restrict__ ddst,
                                const float* __restrict__ bias,
                                const float* __restrict__ res,
                                float* __restrict__ out, int n) {
  int i = blockIdx.x * blockDim.x + threadIdx.x;
  if (i >= n) return;
  const int row = i >> 7;                 // FDIM == 128
  const int col = i & (FDIM - 1);
  float v = agg[i] * ddst[row] + bias[col];
  if constexpr (ADD_RES) v += res[i];
  if constexpr (RELU) v = fmaxf(v, 0.0f);
  out[i] = v;
}

// ---------------------------------------------------------------------------
extern "C" void kernel_launch(void* const* d_in, const int* in_sizes, int n_in,
                              void* d_out, int out_size, void* d_ws, size_t ws_size,
                              hipStream_t stream) {
  // setup_inputs() order
  const float* x   = (const float*)d_in[0];
  const int*   src = (const int*)  d_in[1];
  const int*   dst = (const int*)  d_in[2];
  const float* W1  = (const float*)d_in[3];
  const float* b1  = (const float*)d_in[4];
  const float* W2  = (const float*)d_in[5];
  const float* b2  = (const float*)d_in[6];
  const float* W3  = (const float*)d_in[7];
  const float* b3  = (const float*)d_in[8];
  const float* Wr  = (const float*)d_in[9];
  const float* br  = (const float*)d_in[10];
  const float* Wo  = (const float*)d_in[11];
  const float* bo  = (const float*)d_in[12];
  float* out = (float*)d_out;

  // workspace layout (floats); total ~= 103.1 MB
  float* ws    = (float*)d_ws;
  float* dsrc  = ws;                              // [N]
  float* ddst  = dsrc + NODES;                    // [N]
  float* RES   = ddst + NODES;                    // [N,128] residual
  float* ABUF  = RES  + (size_t)NODES * FDIM;     // [N,128] layer activations
  float* HBUF  = ABUF + (size_t)NODES * FDIM;     // [N,128] x@W staging
  float* AGG   = HBUF + (size_t)NODES * FDIM;     // [N,128] scatter target
  float* WTr   = AGG  + (size_t)NODES * FDIM;     // [128,128] Wr^T
  float* WT1   = WTr  + FDIM * FDIM;              // [128,128] W1^T
  float* WT2   = WT1  + FDIM * FDIM;              // [128,128] W2^T
  float* WT3   = WT2  + FDIM * FDIM;              // [128,128] W3^T
  float* WTo   = WT3  + FDIM * FDIM;              // [64,128]  Wo^T

  const int NF       = NODES * FDIM;
  const int zb_norm  = (2 * NODES + 255) / 256;
  const int zb_feat  = (NF + 255) / 256;
  const int eb       = (EDGES + 255) / 256;
  const int wb       = (FDIM * FDIM + 255) / 256; // 64 blocks for 128x128
  const int wbo      = (FDIM * CDIM + 255) / 256;
  const int mblocks  = NODES / (16 * MTILE);      // 625, exact
  const dim3 gemmH_blk(256);                      // 8 waves -> 8 col tiles (128)
  const dim3 gemmC_blk(128);                      // 4 waves -> 4 col tiles (64)
  const dim3 scat_grid(4096), scat_blk(256);      // 32768 waves over 800k edges

  // ---- degree norms (deg accumulators live in dsrc/ddst, transformed in place)
  zero_f32_kernel<<<zb_norm, 256, 0, stream>>>(dsrc, 2 * NODES);
  degree_kernel<<<eb, 256, 0, stream>>>(src, dst, dsrc, ddst, EDGES);
  norm_kernel<<<(NODES + 255) / 256, 256, 0, stream>>>(dsrc, ddst, NODES);

  // ---- weight transposes (once per call; 64KB each)
  transpose_w_kernel<<<wb,  256, 0, stream>>>(Wr, WTr, FDIM, FDIM);
  transpose_w_kernel<<<wb,  256, 0, stream>>>(W1, WT1, FDIM, FDIM);
  transpose_w_kernel<<<wb,  256, 0, stream>>>(W2, WT2, FDIM, FDIM);
  transpose_w_kernel<<<wb,  256, 0, stream>>>(W3, WT3, FDIM, FDIM);
  transpose_w_kernel<<<wbo, 256, 0, stream>>>(Wo, WTo, FDIM, CDIM);

  // ---- residual: RES = x @ Wr + br
  gemm_k128_wmma<FDIM, false, true>
      <<<mblocks, gemmH_blk, 0, stream>>>(x, WTr, br, nullptr, RES);

  // ---- layer 1: HBUF = diag(dsrc) x @ W1 ; AGG = scatter ; ABUF = relu(...)
  gemm_k128_wmma<FDIM, true, false>
      <<<mblocks, gemmH_blk, 0, stream>>>(x, WT1, nullptr, dsrc, HBUF);
  zero_f32_kernel<<<zb_feat, 256, 0, stream>>>(AGG, NF);
  scatter_edges_kernel<<<scat_grid, scat_blk, 0, stream>>>(HBUF, src, dst, AGG, EDGES);
  epilogue_kernel<true, false>
      <<<zb_feat, 256, 0, stream>>>(AGG, ddst, b1, nullptr, ABUF, NF);

  // ---- layer 2
  gemm_k128_wmma<FDIM, true, false>
      <<<mblocks, gemmH_blk, 0, stream>>>(ABUF, WT2, nullptr, dsrc, HBUF);
  zero_f32_kernel<<<zb_feat, 256, 0, stream>>>(AGG, NF);
  scatter_edges_kernel<<<scat_grid, scat_blk, 0, stream>>>(HBUF, src, dst, AGG, EDGES);
  epilogue_kernel<true, false>
      <<<zb_feat, 256, 0, stream>>>(AGG, ddst, b2, nullptr, ABUF, NF);

  // ---- layer 3 (no relu inside conv) + residual + relu, fused
  gemm_k128_wmma<FDIM, true, false>
      <<<mblocks, gemmH_blk, 0, stream>>>(ABUF, WT3, nullptr, dsrc, HBUF);
  zero_f32_kernel<<<zb_feat, 256, 0, stream>>>(AGG, NF);
  scatter_edges_kernel<<<scat_grid, scat_blk, 0, stream>>>(HBUF, src, dst, AGG, EDGES);
  epilogue_kernel<true, true>
      <<<zb_feat, 256, 0, stream>>>(AGG, ddst, b3, RES, ABUF, NF);

  // ---- output projection: out = ABUF @ Wo + bo  (N x 64)
  gemm_k128_wmma<CDIM, false, true>
      <<<mblocks, gemmC_blk, 0, stream>>>(ABUF, WTo, bo, nullptr, out);
}